// GNNDecoder_7756710937213
// MI455X (gfx1250) — compile-verified
//
#include <hip/hip_runtime.h>
#include <hip/hip_bf16.h>

// GNN decoder: B=16 batches, N=256 nodes, LAT=64, HD=128, L=3 GCN layers.
// f32 end-to-end; GEMMs on v_wmma_f32_16x16x4_f32; all LDS staging goes
// through the CDNA5 async-LDS path (global_load_async_to_lds_b128 +
// s_wait_asynccnt), per CDNA5 ISA.

#define Bb  16
#define Nn  256
#define LAT 64
#define HD  128
#define Ll  3

typedef __attribute__((ext_vector_type(2))) float v2f;
typedef __attribute__((ext_vector_type(8))) float v8f;

// low 32 bits of a generic pointer to __shared__ == LDS byte offset
__device__ __forceinline__ unsigned lds_addr(const void* p) {
    return (unsigned)(unsigned long long)p;
}

// async copy of 16 bytes per lane: from global address gaddr to LDS byte offset ldsByte
__device__ __forceinline__ void async_b128(unsigned ldsByte, const float* gaddr) {
    asm volatile("global_load_async_to_lds_b128 %0, %1, off"
                 :: "v"(ldsByte), "v"(gaddr) : "memory");
}

__device__ __forceinline__ void wait_async0() {
    asm volatile("s_wait_asynccnt 0x0" ::: "memory");
}

// ---------------------------------------------------------------------------
// 1) h[b,n,:] = z[b,:] @ We + be   (broadcast over n)
// ---------------------------------------------------------------------------
__global__ __launch_bounds__(HD) void embed_kernel(
    const float* __restrict__ z, const float* __restrict__ We,
    const float* __restrict__ be, float* __restrict__ h)
{
    const int b = blockIdx.x;       // 0..15
    const int d = threadIdx.x;      // 0..127
    float acc = be[d];
    #pragma unroll 8
    for (int k = 0; k < LAT; ++k)
        acc = fmaf(z[b * LAT + k], We[k * HD + d], acc);
    float* dst = h + (size_t)b * Nn * HD + d;
    for (int n = 0; n < Nn; ++n)
        dst[(size_t)n * HD] = acc;
}

// ---------------------------------------------------------------------------
// 2) C(rows x 128) = A(rows x 128) @ W(128 x 128) [+ bias], f32 WMMA.
//    One wave per 16x16 tile, 4 waves/block sharing one 16-col W panel that
//    is async-staged into LDS (8 KB) once per block.
//    A operand (ISA 7.12.2, 32-bit A 16x4): lane l -> M=l%16,
//      VGPR0/1 hold K = 2*(l/16) + {0,1}.
//    B operand (4x16): lane l -> N=l%16, VGPR0/1 hold rows K = 2*(l/16)+{0,1}.
//    C/D: VGPR r, lanes 0-15 -> row r, lanes 16-31 -> row r+8.
// ---------------------------------------------------------------------------
__global__ __launch_bounds__(128) void gemm128_wmma(
    const float* __restrict__ A, const float* __restrict__ W,
    const float* __restrict__ bias, float* __restrict__ C, int rows)
{
    __shared__ float Wl[HD * 16];                  // W[k][n0..n0+15], row-major

    const int lane = threadIdx.x & 31;
    const int wave = threadIdx.x >> 5;
    const int m0 = (blockIdx.y * 4 + wave) * 16;   // row tile base (grid exact)
    const int n0 = blockIdx.x * 16;                // col tile base

    // async-stage the 128x16 W panel: 512 b128 chunks, 4 per thread
    for (int q = threadIdx.x; q < 512; q += 128) {
        const int row = q >> 2;                    // K index 0..127
        const int c4  = (q & 3) << 2;              // col within tile, x4
        async_b128(lds_addr(&Wl[row * 16 + c4]), W + (size_t)row * HD + n0 + c4);
    }
    wait_async0();
    __syncthreads();

    const int mn   = lane & 15;                    // M (for A) / N (for B)
    const int half = lane >> 4;                    // K-pair selector

    const float* aptr = A + (size_t)(m0 + mn) * HD + 2 * half;
    const float* wl   = &Wl[2 * half * 16 + mn];

    v8f acc = {};
    #pragma unroll 4
    for (int k = 0; k < HD; k += 4) {
        v2f av = *(const v2f*)(aptr + k);          // A[m, k+2h], A[m, k+2h+1]
        v2f bv;
        bv.x = wl[k * 16];                         // W[k+2h,   n]
        bv.y = wl[k * 16 + 16];                    // W[k+2h+1, n]
        acc = __builtin_amdgcn_wmma_f32_16x16x4_f32(
            /*neg_a=*/false, av, /*neg_b=*/false, bv,
            /*c_mod=*/(short)0, acc, /*reuse_a=*/false, /*reuse_b=*/false);
    }

    const int col = n0 + mn;
    const float bvv = bias ? bias[col] : 0.0f;
    #pragma unroll
    for (int r = 0; r < 8; ++r) {
        const int row = m0 + r + 8 * half;
        C[(size_t)row * HD + col] = acc[r] + bvv;
    }
    (void)rows;
}

// ---------------------------------------------------------------------------
// 3) GCN post-processing. Batch 0: inclusive prefix-sum aggregation with
//    symmetric norm deg[j]=j+1; other batches: identity (deg=1 self loop).
//    Then bias + ReLU. hw -> h.
// ---------------------------------------------------------------------------
__global__ __launch_bounds__(HD) void gcn_post_kernel(
    const float* __restrict__ hw, const float* __restrict__ bg_l,
    float* __restrict__ h)
{
    const int b = blockIdx.x;       // 0..15
    const int d = threadIdx.x;      // 0..127
    const float bias = bg_l[d];
    if (b == 0) {
        float running = 0.0f;       // sum_{i<n} dinv[i] * hw[0,i,d]
        for (int n = 0; n < Nn; ++n) {
            const float v    = hw[(size_t)n * HD + d];
            const float dinv = 1.0f / sqrtf((float)(n + 1));
            const float agg  = dinv * running + v * dinv * dinv;
            running = fmaf(v, dinv, running);
            const float o = agg + bias;
            h[(size_t)n * HD + d] = o > 0.0f ? o : 0.0f;
        }
    } else {
        const float* src = hw + (size_t)b * Nn * HD;
        float*       dst = h  + (size_t)b * Nn * HD;
        for (int n = 0; n < Nn; ++n) {
            const float o = src[(size_t)n * HD + d] + bias;
            dst[(size_t)n * HD + d] = o > 0.0f ? o : 0.0f;
        }
    }
}

// ---------------------------------------------------------------------------
// 4) Pairwise edge predictor over the upper triangle.
//    logits[i,j] = sum_d relu(Aq[i,d] + Bk[j,d]) * W2[d] + b2 ; P = sigmoid.
//    out[b,i,j] = P (i<j), mirrored to out[b,j,i]; diagonal = 0.
//    grid.x enumerates the 136 upper-triangular 16x16 tile pairs, grid.y = b.
//    Tiles are async-staged into LDS.
// ---------------------------------------------------------------------------
#define TPAD 132   // LDS row stride: 132*4 % 16 == 0 (b128-aligned rows),
                   // 132 % 64 == 4 -> Bs[jj*TPAD+d] conflict-free over jj

__global__ __launch_bounds__(256) void pairwise_kernel(
    const float* __restrict__ Aq, const float* __restrict__ Bk,
    const float* __restrict__ W2, const float* __restrict__ b2,
    float* __restrict__ out)
{
    __shared__ float As[16 * TPAD];
    __shared__ float Bs[16 * TPAD];
    __shared__ float W2s[HD];

    const int b   = blockIdx.y;
    const int tid = threadIdx.x;

    // map linear tile id -> (ti, tj) with ti <= tj over 16x16 node tiles
    int rem = blockIdx.x, ti = 0;
    while (rem >= 16 - ti) { rem -= 16 - ti; ++ti; }
    const int tj = ti + rem;

    const float* ag = Aq + ((size_t)b * Nn + ti * 16) * HD;
    const float* bg = Bk + ((size_t)b * Nn + tj * 16) * HD;

    // 512 b128 chunks per 16x128 tile; 2 per thread per tile
    #pragma unroll
    for (int q = tid; q < 512; q += 256) {
        const int row = q >> 5;                    // 0..15
        const int c4  = (q & 31) << 2;             // 0,4,..,124
        async_b128(lds_addr(&As[row * TPAD + c4]), ag + row * HD + c4);
        async_b128(lds_addr(&Bs[row * TPAD + c4]), bg + row * HD + c4);
    }
    if (tid < 32) async_b128(lds_addr(&W2s[tid * 4]), W2 + tid * 4);
    wait_async0();
    __syncthreads();

    const int ii = tid >> 4;        // 0..15 row within tile
    const int jj = tid & 15;        // 0..15 col within tile
    const int i  = ti * 16 + ii;
    const int j  = tj * 16 + jj;

    float acc = b2[0];
    const float* ar = &As[ii * TPAD];
    const float* br = &Bs[jj * TPAD];
    #pragma unroll 8
    for (int d = 0; d < HD; ++d) {
        float s = fmaxf(ar[d] + br[d], 0.0f);
        acc = fmaf(s, W2s[d], acc);
    }

    float* ob = out + (size_t)b * Nn * Nn;
    if (i < j) {
        const float p = 1.0f / (1.0f + expf(-acc));
        ob[(size_t)i * Nn + j] = p;
        ob[(size_t)j * Nn + i] = p;
    } else if (i == j) {
        ob[(size_t)i * Nn + i] = 0.0f;   // diagonal stays zero
    }
    // i > j happens only inside diagonal tiles; the mirrored thread writes it.
}

// ---------------------------------------------------------------------------
extern "C" void kernel_launch(void* const* d_in, const int* in_sizes, int n_in,
                              void* d_out, int out_size, void* d_ws, size_t ws_size,
                              hipStream_t stream)
{
    const float* z   = (const float*)d_in[0];
    const float* We  = (const float*)d_in[1];
    const float* be  = (const float*)d_in[2];
    const float* Wg  = (const float*)d_in[3];
    const float* bgp = (const float*)d_in[4];
    const float* W1a = (const float*)d_in[5];
    const float* W1b = (const float*)d_in[6];
    const float* b1  = (const float*)d_in[7];
    const float* W2  = (const float*)d_in[8];
    const float* b2  = (const float*)d_in[9];
    float* out = (float*)d_out;

    const size_t HSZ = (size_t)Bb * Nn * HD;       // 524288 floats
    float* h  = (float*)d_ws;
    float* t0 = h  + HSZ;
    float* t1 = t0 + HSZ;

    const int rows = Bb * Nn;                      // 4096
    const dim3 gemmGrid(HD / 16, rows / 64);       // (8, 64), 4 waves/block

    // node embedding (broadcast)
    embed_kernel<<<Bb, HD, 0, stream>>>(z, We, be, h);

    // 3 GCN layers: WMMA gemm, then aggregation + bias + relu
    for (int l = 0; l < Ll; ++l) {
        gemm128_wmma<<<gemmGrid, 128, 0, stream>>>(h, Wg + (size_t)l * HD * HD,
                                                   nullptr, t0, rows);
        gcn_post_kernel<<<Bb, HD, 0, stream>>>(t0, bgp + (size_t)l * HD, h);
    }

    // edge-MLP projections: A = h@W1a + b1 ; Bm = h@W1b
    gemm128_wmma<<<gemmGrid, 128, 0, stream>>>(h, W1a, b1,      t0, rows);
    gemm128_wmma<<<gemmGrid, 128, 0, stream>>>(h, W1b, nullptr, t1, rows);

    // pairwise sigmoid predictor over the upper triangle (+mirror, zero diag)
    pairwise_kernel<<<dim3(136, Bb), 256, 0, stream>>>(t0, t1, W2, b2, out);

    (void)in_sizes; (void)n_in; (void)out_size; (void)ws_size;
}